// GAT_51634096833138
// MI455X (gfx1250) — compile-verified
//
#include <hip/hip_runtime.h>
#include <math.h>

#define H 128
#define WARP 32
#define APAD 132   // LDS row stride (floats); 128 would alias all rows to same banks

typedef __attribute__((ext_vector_type(2))) float v2f;
typedef __attribute__((ext_vector_type(8))) float v8f;

// Guaranteed-hardware fp32 atomic add (no return -> STOREcnt), device scope so the
// RMW executes in the coherent L2 where out/denom live.
__device__ __forceinline__ void atomic_fadd_f32(float* p, float v) {
    asm volatile("global_atomic_add_f32 %0, %1, off scope:SCOPE_DEV"
                 :: "v"(p), "v"(v) : "memory");
}

// Order-preserving float->uint key so we can use hardware integer atomic max.
__device__ __forceinline__ unsigned f2key(float f) {
    unsigned u = __float_as_uint(f);
    return (u & 0x80000000u) ? ~u : (u | 0x80000000u);
}
__device__ __forceinline__ float key2f(unsigned k) {
    return (k & 0x80000000u) ? __uint_as_float(k & 0x7fffffffu)
                             : __uint_as_float(~k);
}

// --- Kernel 1: per-node attention scores si = x.a_i, sj = x.a_j (wave per node)
__global__ void gat_scores(const float* __restrict__ x,
                           const float* __restrict__ ai,
                           const float* __restrict__ aj,
                           float* __restrict__ si, float* __restrict__ sj, int n) {
    int wave = (int)((blockIdx.x * blockDim.x + threadIdx.x) / WARP);
    int lane = threadIdx.x & (WARP - 1);
    if (wave >= n) return;
    float4 xv = ((const float4*)(x + (size_t)wave * H))[lane];
    float4 av = ((const float4*)ai)[lane];
    float4 bv = ((const float4*)aj)[lane];
    float di = xv.x * av.x + xv.y * av.y + xv.z * av.z + xv.w * av.w;
    float dj = xv.x * bv.x + xv.y * bv.y + xv.z * bv.z + xv.w * bv.w;
    #pragma unroll
    for (int off = 16; off; off >>= 1) {
        di += __shfl_xor(di, off, WARP);
        dj += __shfl_xor(dj, off, WARP);
    }
    if (lane == 0) { si[wave] = di; sj[wave] = dj; }
}

// --- Kernel 2: m = x + x @ W via V_WMMA_F32_16X16X4_F32
// Block = 256 threads = 8 waves. Block computes 16 rows; wave w computes cols [16w,16w+16).
// A (16x4 f32): lanes 0-15 = M rows, VGPR0/1 = K,K+1; lanes 16-31 = K+2,K+3.
// B (4x16 f32): VGPR0 = row K (lanes 0-15) / row K+2 (lanes 16-31); VGPR1 = K+1 / K+3.
__global__ void gat_gemm(const float* __restrict__ x,
                         const float* __restrict__ W,
                         float* __restrict__ m, int n) {
    __shared__ float As[16 * APAD];
    int t  = threadIdx.x;
    int r0 = blockIdx.x * 16;

    // Stage 16x128 fp32 tile of x into LDS (512 float4 loads, 2 per thread)
    #pragma unroll
    for (int i = 0; i < 2; ++i) {
        int idx  = t + i * 256;          // 0..511
        int row  = idx >> 5;             // 32 float4 per row
        int col4 = idx & 31;
        int gr = r0 + row; if (gr >= n) gr = n - 1;   // clamp (no EXEC divergence)
        float4 v = ((const float4*)(x + (size_t)gr * H))[col4];
        float* dst = &As[row * APAD + col4 * 4];
        dst[0] = v.x; dst[1] = v.y; dst[2] = v.z; dst[3] = v.w;
    }
    __syncthreads();

    int wave = t >> 5;                 // 0..7  -> output column tile
    int lane = t & 31;
    int nn   = wave * 16 + (lane & 15);    // global output column
    int mrow = lane & 15;                  // A row held by this lane
    int koff = (lane >> 4) << 1;           // 0 for lanes 0-15, 2 for lanes 16-31

    v8f acc = {};
    #pragma unroll
    for (int kk = 0; kk < H; kk += 4) {
        int k = kk + koff;
        v2f a, b;
        a.x = As[mrow * APAD + k];
        a.y = As[mrow * APAD + k + 1];
        b.x = W[(size_t)k * H + nn];
        b.y = W[(size_t)(k + 1) * H + nn];
        acc = __builtin_amdgcn_wmma_f32_16x16x4_f32(
            /*neg_a=*/false, a, /*neg_b=*/false, b,
            /*c_mod=*/(short)0, acc, /*reuse_a=*/false, /*reuse_b=*/false);
    }

    // Epilogue: D VGPR v holds row (v + 8*(lane>=16)), col = lane&15 within tile.
    int hi = (lane >> 4) * 8;
    #pragma unroll
    for (int v = 0; v < 8; ++v) {
        int row = v + hi;
        int gr  = r0 + row;
        if (gr < n) {
            m[(size_t)gr * H + nn] = acc[v] + As[row * APAD + nn];
        }
    }
}

// --- Kernel 3: edge scores + segment max (atomic uint-key max)
__global__ void gat_edge_score(const float* __restrict__ si, const float* __restrict__ sj,
                               const int* __restrict__ edges, int E_,
                               float* __restrict__ e_buf, unsigned* __restrict__ segmax) {
    int e = blockIdx.x * blockDim.x + threadIdx.x;
    if (e >= E_) return;
    int j = edges[e];        // edge_index[0]: output node
    int i = edges[E_ + e];   // edge_index[1]: softmax segment / source
    float s  = si[i] + sj[j];
    float lr = s > 0.0f ? s : 0.01f * s;   // leaky_relu(0.01)
    e_buf[e] = lr;
    atomicMax(&segmax[i], f2key(lr));
}

// --- Kernel 4: exp(e - segmax) and segment denom
__global__ void gat_edge_exp(const int* __restrict__ edges, int E_,
                             float* __restrict__ e_buf,
                             const unsigned* __restrict__ segmax,
                             float* __restrict__ denom) {
    int e = blockIdx.x * blockDim.x + threadIdx.x;
    if (e >= E_) return;
    int i = edges[E_ + e];
    float ex = expf(e_buf[e] - key2f(segmax[i]));
    e_buf[e] = ex;
    atomic_fadd_f32(&denom[i], ex);
}

// --- Kernel 5: out[j] += alpha * m[i]  (wave per edge; 4 coalesced hw atomics/lane)
__global__ void gat_scatter(const int* __restrict__ edges, int E_,
                            const float* __restrict__ e_buf,
                            const float* __restrict__ denom,
                            const float* __restrict__ m,
                            float* __restrict__ out) {
    int wave = (int)((blockIdx.x * blockDim.x + threadIdx.x) >> 5);
    int lane = threadIdx.x & 31;
    if (wave >= E_) return;
    int j = edges[wave];
    int i = edges[E_ + wave];
    float alpha = e_buf[wave] / denom[i];
    const float* mi = m + (size_t)i * H;
    float* oj = out + (size_t)j * H;
    #pragma unroll
    for (int q = 0; q < 4; ++q) {
        int c = lane + q * 32;
        atomic_fadd_f32(&oj[c], alpha * mi[c]);
    }
}

// --- Kernel 6: exact-erf GELU, in place on the accumulated output
__global__ void gat_gelu(float* __restrict__ out, long long total) {
    long long idx = (long long)blockIdx.x * blockDim.x + threadIdx.x;
    if (idx >= total) return;
    float v = out[idx];
    out[idx] = 0.5f * v * (1.0f + erff(v * 0.70710678118654752f));
}

extern "C" void kernel_launch(void* const* d_in, const int* in_sizes, int n_in,
                              void* d_out, int out_size, void* d_ws, size_t ws_size,
                              hipStream_t stream) {
    const float* x     = (const float*)d_in[0];   // [N,128]
    const int*   edges = (const int*)d_in[1];     // [2,E]
    const float* ai    = (const float*)d_in[2];   // [128]
    const float* aj    = (const float*)d_in[3];   // [128]
    const float* W     = (const float*)d_in[4];   // [128,128]
    float* out = (float*)d_out;

    int N = in_sizes[0] / H;   // 50000
    int E = in_sizes[1] / 2;   // 800000

    // Workspace layout (floats): m[N*H] | si[N] | sj[N] | e_buf[E] | segmax[N] | denom[N]
    float*    m_ws   = (float*)d_ws;
    float*    si     = m_ws + (size_t)N * H;
    float*    sj     = si + N;
    float*    e_buf  = sj + N;
    unsigned* segmax = (unsigned*)(e_buf + E);
    float*    denom  = (float*)(segmax + N);

    // Zero segmax(key 0 acts as -inf) + denom (contiguous), and the atomic output buffer.
    hipMemsetAsync(segmax, 0, (size_t)N * 2 * sizeof(float), stream);
    hipMemsetAsync(d_out, 0, (size_t)N * H * sizeof(float), stream);

    gat_scores<<<(N + 7) / 8, 256, 0, stream>>>(x, ai, aj, si, sj, N);
    gat_gemm<<<(N + 15) / 16, 256, 0, stream>>>(x, W, m_ws, N);

    int eb = (E + 255) / 256;
    gat_edge_score<<<eb, 256, 0, stream>>>(si, sj, edges, E, e_buf, segmax);
    gat_edge_exp<<<eb, 256, 0, stream>>>(edges, E, e_buf, segmax, denom);
    gat_scatter<<<(E + 7) / 8, 256, 0, stream>>>(edges, E, e_buf, denom, m_ws, out);

    long long total = (long long)N * H;
    gat_gelu<<<(int)((total + 255) / 256), 256, 0, stream>>>(out, total);
}